// Top2Gating_45011257262411
// MI455X (gfx1250) — compile-verified
//
#include <hip/hip_runtime.h>
#include <hip/hip_bf16.h>

typedef __attribute__((ext_vector_type(2))) float v2f;
typedef __attribute__((ext_vector_type(4))) float v4f;
typedef __attribute__((ext_vector_type(8))) float v8f;

#define BB   8
#define NN   2048
#define DD   1024
#define EE   8
#define EP   16           // padded expert columns for branchless B fragments
#define CAP  512          // max(4, min(2048, int(2048*2.0/8)))
#define TOK  (BB*NN)      // 16384
#define ECAP (EE*CAP)     // 4096

// ---------------------------------------------------------------------------
// Kernel 0: zero the small accumulators (gsum) in workspace
// ---------------------------------------------------------------------------
__global__ void k_zero(float* gsum) {
    if (threadIdx.x < BB*EE) gsum[threadIdx.x] = 0.0f;
}

// ---------------------------------------------------------------------------
// Kernel 1: gating via V_WMMA_F32_16X16X4_F32.
// Block = 256 threads = 8 waves; each wave owns a 16-token tile.
// A frag (16x4 f32): lane<16 -> row=lane, K=k..k+1 ; lane>=16 -> row=lane-16, K=k+2..k+3
// B frag (4x16 f32): lane<16 -> col=lane, K=k..k+1 ; lane>=16 -> col=lane-16, K=k+2..k+3
// C/D (16x16 f32):   VGPR r: lanes 0-15 -> M=r, N=lane ; lanes 16-31 -> M=r+8, N=lane-16
// B is staged in LDS padded to 16 columns (cols 8..15 = 0) so the inner loop
// has no divergence: global_load_b64 + ds_load_2addr_b32 + v_wmma per step.
// ---------------------------------------------------------------------------
__global__ void k_gating(const float* __restrict__ x,
                         const float* __restrict__ w,
                         int* __restrict__ idx1, int* __restrict__ idx2,
                         float* __restrict__ g1,  float* __restrict__ g2,
                         float* __restrict__ gsum)
{
    __shared__ float sW[DD * EP];          // 64 KB: gating matrix, zero-padded
    __shared__ float sLog[8 * 16 * 17];    // per-wave 16x16 logits, padded stride 17
    __shared__ float sGsum[EE];

    const int tid = threadIdx.x;

    // preload w (zero-pad columns 8..15)
    for (int i = tid; i < DD * EE; i += 256) {
        const int d = i >> 3, c = i & 7;
        sW[d * EP + c]      = w[i];
        sW[d * EP + 8 + c]  = 0.0f;
    }
    if (tid < EE) sGsum[tid] = 0.0f;
    __syncthreads();

    const int wave = tid >> 5;
    const int lane = tid & 31;
    const int hf   = lane >> 4;            // 0 or 1 (lane half)
    const int ln   = lane & 15;

    const int tile0   = (blockIdx.x * 8 + wave) * 16;   // first token of this wave
    const float* xrow = x + (size_t)(tile0 + ln) * DD + 2 * hf;
    const float* wcol = sW + 2 * hf * EP + ln;          // K offset for this lane half

    v8f acc = {};
#pragma unroll 8
    for (int k = 0; k < DD; k += 4) {
        v2f a = *(const v2f*)(xrow + k);                // 8B aligned
        v2f b;
        b.x = wcol[k * EP];
        b.y = wcol[k * EP + EP];
        acc = __builtin_amdgcn_wmma_f32_16x16x4_f32(
            /*neg_a=*/false, a, /*neg_b=*/false, b,
            /*c_mod=*/(short)0, acc, /*reuse_a=*/false, /*reuse_b=*/false);
    }

    // stage C tile to LDS: token t = r + 8*hf, expert = ln
    float* myLog = sLog + wave * 272;
#pragma unroll
    for (int r = 0; r < 8; ++r)
        myLog[(r + 8 * hf) * 17 + ln] = acc[r];
    __syncthreads();

    // one thread per token: softmax over 8 experts, top-1 / top-2, normalize
    if (tid < 128) {
        const float* lg = sLog + (tid >> 4) * 272 + (tid & 15) * 17;
        float p[EE];
        float mx = lg[0];
#pragma unroll
        for (int e = 0; e < EE; ++e) { p[e] = lg[e]; mx = fmaxf(mx, p[e]); }
        float s = 0.0f;
#pragma unroll
        for (int e = 0; e < EE; ++e) { p[e] = expf(p[e] - mx); s += p[e]; }
        const float inv = 1.0f / s;
#pragma unroll
        for (int e = 0; e < EE; ++e) p[e] *= inv;

        int i1 = 0; float m1 = p[0];
#pragma unroll
        for (int e = 1; e < EE; ++e) if (p[e] > m1) { m1 = p[e]; i1 = e; }
        int i2 = (i1 == 0) ? 1 : 0; float m2 = -1.0f;
#pragma unroll
        for (int e = 0; e < EE; ++e) if (e != i1 && p[e] > m2) { m2 = p[e]; i2 = e; }

        const float denom = m1 + m2 + 1e-9f;
        const int tok = blockIdx.x * 128 + tid;
        idx1[tok] = i1; idx2[tok] = i2;
        g1[tok] = m1 / denom; g2[tok] = m2 / denom;

#pragma unroll
        for (int e = 0; e < EE; ++e) atomicAdd(&sGsum[e], p[e]);
    }
    __syncthreads();
    if (tid < EE) atomicAdd(&gsum[(blockIdx.x >> 4) * EE + tid], sGsum[tid]);
}

// ---------------------------------------------------------------------------
// Kernel 2: per-batch capacity scan (one block per batch, 256 thr = 8 waves).
// Wave32 ballots for per-expert exclusive prefix counts; LDS for cross-wave.
// ---------------------------------------------------------------------------
__global__ void k_scan(const int* __restrict__ idx1, const int* __restrict__ idx2,
                       const float* __restrict__ g1, const float* __restrict__ g2,
                       int* __restrict__ pos1, int* __restrict__ pos2,
                       float* __restrict__ g1e, float* __restrict__ g2e,
                       float* __restrict__ cnt1)
{
    __shared__ int base[EE];
    __shared__ int waveTot[8][EE];

    const int b    = blockIdx.x;
    const int tid  = threadIdx.x;
    const int wave = tid >> 5;
    const int lane = tid & 31;
    const unsigned lt = (1u << lane) - 1u;
    const int tokBase = b * NN;

    if (tid < EE) base[tid] = 0;
    __syncthreads();

    // ---- pass 1: top-1 positions ----
    for (int c = 0; c < NN; c += 256) {
        const int tok = tokBase + c + tid;
        const int e1  = idx1[tok];
        int myPre = 0;
#pragma unroll
        for (int e = 0; e < EE; ++e) {
            unsigned m = (unsigned)__ballot(e1 == e);
            if (lane == e) waveTot[wave][e] = __popc(m);
            if (e1 == e)   myPre = __popc(m & lt);
        }
        __syncthreads();
        int off = 0;
        for (int w2 = 0; w2 < wave; ++w2) off += waveTot[w2][e1];
        const int p = base[e1] + off + myPre;
        pos1[tok] = p;
        g1e[tok]  = (p < CAP) ? g1[tok] : 0.0f;
        __syncthreads();
        if (tid < EE) {
            int t = 0;
            for (int w2 = 0; w2 < 8; ++w2) t += waveTot[w2][tid];
            base[tid] += t;
        }
        __syncthreads();
    }

    // pre-capacity counts feed the loss; clamped counts offset pass 2
    if (tid < EE) {
        cnt1[b * EE + tid] = (float)base[tid];
        base[tid] = min(base[tid], CAP);
    }
    __syncthreads();

    // ---- pass 2: top-2 positions (offset by clamped top-1 occupancy) ----
    for (int c = 0; c < NN; c += 256) {
        const int tok = tokBase + c + tid;
        const int e2  = idx2[tok];
        int myPre = 0;
#pragma unroll
        for (int e = 0; e < EE; ++e) {
            unsigned m = (unsigned)__ballot(e2 == e);
            if (lane == e) waveTot[wave][e] = __popc(m);
            if (e2 == e)   myPre = __popc(m & lt);
        }
        __syncthreads();
        int off = 0;
        for (int w2 = 0; w2 < wave; ++w2) off += waveTot[w2][e2];
        const int p = base[e2] + off + myPre;
        pos2[tok] = p;
        g2e[tok]  = (p < CAP) ? g2[tok] : 0.0f;
        __syncthreads();
        if (tid < EE) {
            int t = 0;
            for (int w2 = 0; w2 < 8; ++w2) t += waveTot[w2][tid];
            base[tid] += t;
        }
        __syncthreads();
    }
}

// ---------------------------------------------------------------------------
// Kernel 3: write dispatch + combine. One block per token; each tensor row is
// e*cap = 4096 floats -> 4 non-temporal b128 stores per thread per tensor.
// 537 MB of writes >> 192 MB L2, so bypass with non-temporal hints.
// ---------------------------------------------------------------------------
__global__ void k_fill(const int* __restrict__ idx1, const int* __restrict__ pos1,
                       const float* __restrict__ g1e,
                       const int* __restrict__ idx2, const int* __restrict__ pos2,
                       const float* __restrict__ g2e,
                       float* __restrict__ out)
{
    const int tok = blockIdx.x;
    const int tid = threadIdx.x;

    const float v1 = g1e[tok], v2 = g2e[tok];
    const int nz1 = (v1 != 0.0f) ? idx1[tok] * CAP + pos1[tok] : -1;
    const int nz2 = (v2 != 0.0f) ? idx2[tok] * CAP + pos2[tok] : -1;

    float* disp = out + (size_t)tok * ECAP;
    float* comb = out + (size_t)TOK * ECAP + (size_t)tok * ECAP;

#pragma unroll
    for (int r = 0; r < 4; ++r) {
        const int f0 = r * 1024 + tid * 4;
        const float c0 = (nz1 == f0 + 0 ? v1 : 0.0f) + (nz2 == f0 + 0 ? v2 : 0.0f);
        const float c1 = (nz1 == f0 + 1 ? v1 : 0.0f) + (nz2 == f0 + 1 ? v2 : 0.0f);
        const float c2 = (nz1 == f0 + 2 ? v1 : 0.0f) + (nz2 == f0 + 2 ? v2 : 0.0f);
        const float c3 = (nz1 == f0 + 3 ? v1 : 0.0f) + (nz2 == f0 + 3 ? v2 : 0.0f);
        v4f vc = { c0, c1, c2, c3 };
        v4f vd = { c0 != 0.0f ? 1.0f : 0.0f, c1 != 0.0f ? 1.0f : 0.0f,
                   c2 != 0.0f ? 1.0f : 0.0f, c3 != 0.0f ? 1.0f : 0.0f };
        __builtin_nontemporal_store(vc, (v4f*)(comb + f0));
        __builtin_nontemporal_store(vd, (v4f*)(disp + f0));
    }
}

// ---------------------------------------------------------------------------
// Kernel 4: loss = e^2/(b*n*e) * sum_{b,e} (cnt1/n)*gsum
// ---------------------------------------------------------------------------
__global__ void k_loss(const float* __restrict__ cnt1, const float* __restrict__ gsum,
                       float* __restrict__ out_loss)
{
    __shared__ float red[64];
    const int t = threadIdx.x;
    red[t] = cnt1[t] * (1.0f / (float)NN) * gsum[t];
    __syncthreads();
    for (int s = 32; s > 0; s >>= 1) {
        if (t < s) red[t] += red[t + s];
        __syncthreads();
    }
    if (t == 0)
        out_loss[0] = red[0] * ((float)(EE * EE) / (float)(BB * NN * EE));
}

// ---------------------------------------------------------------------------
extern "C" void kernel_launch(void* const* d_in, const int* in_sizes, int n_in,
                              void* d_out, int out_size, void* d_ws, size_t ws_size,
                              hipStream_t stream) {
    const float* x = (const float*)d_in[0];   // [8,2048,1024]
    const float* w = (const float*)d_in[1];   // [1024,8]
    float* out = (float*)d_out;               // dispatch | combine | loss

    char* p = (char*)d_ws;
    int*   idx1 = (int*)p;    p += (size_t)TOK * 4;
    int*   idx2 = (int*)p;    p += (size_t)TOK * 4;
    float* g1   = (float*)p;  p += (size_t)TOK * 4;
    float* g2   = (float*)p;  p += (size_t)TOK * 4;
    int*   pos1 = (int*)p;    p += (size_t)TOK * 4;
    int*   pos2 = (int*)p;    p += (size_t)TOK * 4;
    float* g1e  = (float*)p;  p += (size_t)TOK * 4;
    float* g2e  = (float*)p;  p += (size_t)TOK * 4;
    float* gsum = (float*)p;  p += (size_t)BB * EE * 4;
    float* cnt1 = (float*)p;  p += (size_t)BB * EE * 4;

    k_zero  <<<1, 64, 0, stream>>>(gsum);
    k_gating<<<TOK / 128, 256, 0, stream>>>(x, w, idx1, idx2, g1, g2, gsum);
    k_scan  <<<BB, 256, 0, stream>>>(idx1, idx2, g1, g2, pos1, pos2, g1e, g2e, cnt1);
    k_fill  <<<TOK, 256, 0, stream>>>(idx1, pos1, g1e, idx2, pos2, g2e, out);
    k_loss  <<<1, 64, 0, stream>>>(cnt1, gsum, out + 2 * (size_t)TOK * ECAP);
}